// Attention_51548197486975
// MI455X (gfx1250) — compile-verified
//
#include <hip/hip_runtime.h>

// ---------------------------------------------------------------------------
// Temporal-MoE attention for MI455X (gfx1250), fp32 via V_WMMA_F32_16X16X4_F32
// ---------------------------------------------------------------------------

typedef float v2f __attribute__((ext_vector_type(2)));
typedef float v8f __attribute__((ext_vector_type(8)));

#define DIMC   512
#define HEADS  8
#define HEADD  64
#define TLEN   243
#define NJ     17
#define NBATCH 4
#define NTOK   (NBATCH * TLEN * NJ)          // 16524 token rows of x
#define NSEQ   (NBATCH * HEADS * NJ)         // 544 attention sequences
#define SEQ_ELEMS ((size_t)NSEQ * TLEN * HEADD) // 8,460,288 floats per q/k/v

// workspace layout (in floats)
#define Q_OFF   ((size_t)0)
#define K_OFF   (SEQ_ELEMS)
#define V_OFF   (2 * SEQ_ELEMS)
#define G_OFF   (3 * SEQ_ELEMS)
#define CTX_OFF (3 * SEQ_ELEMS + (size_t)NTOK * 4)

// LDS strides (floats) chosen to avoid 16-way bank conflicts on fragment reads
#define KS_STRIDE 68
#define VS_STRIDE 64
#define SS_STRIDE 260

static __device__ __forceinline__ v8f wmma4(v2f a, v2f b, v8f c) {
  // D = A(16x4, f32) x B(4x16, f32) + C(16x16, f32)
  return __builtin_amdgcn_wmma_f32_16x16x4_f32(false, a, false, b, (short)0, c,
                                               false, false);
}

// ---------------------------------------------------------------------------
// Kernel 1: qkv = x @ qkv_w  (16524 x 512 x 1536), scatter into blocked q/k/v
// layout ws[part][((b*8+h)*17+n)*243 + t][cc]
// ---------------------------------------------------------------------------
__global__ __launch_bounds__(256) void gemm_qkv_kernel(
    const float* __restrict__ X, const float* __restrict__ W,
    float* __restrict__ ws) {
  __shared__ float As[16][128];
  __shared__ float Bs[16][64];
  const int tid  = threadIdx.x;
  const int lane = tid & 31;
  const int wv   = tid >> 5;
  const int m    = lane & 15;
  const int half = lane >> 4;
  const int rowBase = blockIdx.y * 128;
  const int colBase = blockIdx.x * 64;

  v8f zero = {0.f, 0.f, 0.f, 0.f, 0.f, 0.f, 0.f, 0.f};
  v8f acc[4] = {zero, zero, zero, zero};

  for (int k0 = 0; k0 < DIMC; k0 += 16) {
    __syncthreads();
    {  // stage A tile (128 rows x 16 k), transposed As[k][row]
      int r  = tid >> 1;
      int kk = (tid & 1) * 8;
      int rg = rowBase + r;
      float4 v0 = {0.f, 0.f, 0.f, 0.f}, v1 = {0.f, 0.f, 0.f, 0.f};
      if (rg < NTOK) {
        const float4* p = (const float4*)(X + (size_t)rg * DIMC + k0 + kk);
        v0 = p[0];
        v1 = p[1];
      }
      float tmp[8] = {v0.x, v0.y, v0.z, v0.w, v1.x, v1.y, v1.z, v1.w};
#pragma unroll
      for (int j = 0; j < 8; ++j) As[kk + j][r] = tmp[j];
    }
    {  // stage B tile (16 k x 64 cols)
      int kr = tid >> 4;
      int c  = (tid & 15) * 4;
      float4 v = *(const float4*)(W + (size_t)(k0 + kr) * 1536 + colBase + c);
      *(float4*)&Bs[kr][c] = v;
    }
    __syncthreads();
#pragma unroll
    for (int kk = 0; kk < 16; kk += 4) {
      v2f a;
      a.x = As[kk + 2 * half + 0][wv * 16 + m];
      a.y = As[kk + 2 * half + 1][wv * 16 + m];
#pragma unroll
      for (int ct = 0; ct < 4; ++ct) {
        v2f b;
        b.x = Bs[kk + 2 * half + 0][ct * 16 + m];
        b.y = Bs[kk + 2 * half + 1][ct * 16 + m];
        acc[ct] = wmma4(a, b, acc[ct]);
      }
    }
  }

  // scatter epilogue into blocked q/k/v
#pragma unroll
  for (int ct = 0; ct < 4; ++ct) {
    int ncol = colBase + ct * 16 + m;
    int part = ncol >> 9;       // 0=q 1=k 2=v
    int rem  = ncol & 511;
    int h    = rem >> 6;
    int cc   = rem & 63;
    float* base = ws + (size_t)part * SEQ_ELEMS;
#pragma unroll
    for (int r = 0; r < 8; ++r) {
      int rowL = wv * 16 + r + half * 8;
      int rg   = rowBase + rowL;
      if (rg < NTOK) {
        int bt = rg / NJ, n = rg % NJ;
        int b = bt / TLEN, t = bt % TLEN;
        int seq = (b * HEADS + h) * NJ + n;
        base[((size_t)seq * TLEN + t) * HEADD + cc] = acc[ct][r];
      }
    }
  }
}

// ---------------------------------------------------------------------------
// Kernel 2: expert gates = softmax(x @ te_w + te_b), one wave per token
// ---------------------------------------------------------------------------
__global__ __launch_bounds__(256) void gate_kernel(
    const float* __restrict__ X, const float* __restrict__ TW,
    const float* __restrict__ TB, float* __restrict__ gate) {
  const int wv   = threadIdx.x >> 5;
  const int lane = threadIdx.x & 31;
  const int tok  = blockIdx.x * 8 + wv;
  if (tok >= NTOK) return;
  float a0 = 0.f, a1 = 0.f, a2 = 0.f, a3 = 0.f;
#pragma unroll
  for (int i = 0; i < 16; ++i) {
    int c = lane + i * 32;
    float xv = X[(size_t)tok * DIMC + c];
    float4 w = *(const float4*)(TW + c * 4);
    a0 += xv * w.x; a1 += xv * w.y; a2 += xv * w.z; a3 += xv * w.w;
  }
#pragma unroll
  for (int off = 16; off; off >>= 1) {
    a0 += __shfl_xor(a0, off);
    a1 += __shfl_xor(a1, off);
    a2 += __shfl_xor(a2, off);
    a3 += __shfl_xor(a3, off);
  }
  if (lane == 0) {
    a0 += TB[0]; a1 += TB[1]; a2 += TB[2]; a3 += TB[3];
    float mx = fmaxf(fmaxf(a0, a1), fmaxf(a2, a3));
    float e0 = __expf(a0 - mx), e1 = __expf(a1 - mx);
    float e2 = __expf(a2 - mx), e3 = __expf(a3 - mx);
    float inv = 1.f / (e0 + e1 + e2 + e3);
    float4 g = {e0 * inv, e1 * inv, e2 * inv, e3 * inv};
    *(float4*)(gate + (size_t)tok * 4) = g;
  }
}

// ---------------------------------------------------------------------------
// Kernel 3: fused windowed-MoE attention, one block per (b,h,joint) sequence.
// S = QK^T (WMMA) -> hierarchical nested-window softmax x gates -> O = W V.
// ---------------------------------------------------------------------------
extern __shared__ float smem[];

__global__ __launch_bounds__(256) void attn_kernel(float* __restrict__ ws) {
  float* Ks = smem;                        // 256 x KS_STRIDE
  float* Vs = Ks + 256 * KS_STRIDE;        // 256 x VS_STRIDE
  float* Ss = Vs + 256 * VS_STRIDE;        // 8 waves x 16 x SS_STRIDE

  const int seq = blockIdx.x;
  const int n   = seq % NJ;
  const int h   = (seq / NJ) % HEADS;
  const int b   = seq / (NJ * HEADS);
  const int tid = threadIdx.x;
  const int lane = tid & 31;
  const int wv   = tid >> 5;
  const int m    = lane & 15;
  const int half = lane >> 4;

  const float* Qg = ws + Q_OFF + (size_t)seq * TLEN * HEADD;
  const float* Kg = ws + K_OFF + (size_t)seq * TLEN * HEADD;
  const float* Vg = ws + V_OFF + (size_t)seq * TLEN * HEADD;
  const float* Gg = ws + G_OFF;
  float* Cg = ws + CTX_OFF;

  // stage K and V in LDS, zero the pad rows (243..255)
  for (int i = tid; i < 256 * 16; i += 256) {
    int row = i >> 4;
    int c4  = (i & 15) * 4;
    float4 kv = {0.f, 0.f, 0.f, 0.f}, vv = {0.f, 0.f, 0.f, 0.f};
    if (row < TLEN) {
      kv = *(const float4*)(Kg + (size_t)row * HEADD + c4);
      vv = *(const float4*)(Vg + (size_t)row * HEADD + c4);
    }
    *(float4*)(Ks + row * KS_STRIDE + c4) = kv;
    *(float4*)(Vs + row * VS_STRIDE + c4) = vv;
  }
  __syncthreads();

  float* Sw = Ss + wv * 16 * SS_STRIDE;

  for (int strip = wv; strip < 16; strip += 8) {
    // ---- load this strip's Q fragments (16 rows x 64 c) from global ----
    int tq = strip * 16 + m;
    v2f aq[16];
    if (tq < TLEN) {
      const float* qrow = Qg + (size_t)tq * HEADD;
#pragma unroll
      for (int kk = 0; kk < 16; ++kk) {
        int k = kk * 4 + 2 * half;
        aq[kk].x = qrow[k];
        aq[kk].y = qrow[k + 1];
      }
    } else {
#pragma unroll
      for (int kk = 0; kk < 16; ++kk) { aq[kk].x = 0.f; aq[kk].y = 0.f; }
    }

    // ---- S(16x256) = Q_strip (16x64) @ K^T, scaled ----
    for (int j = 0; j < 16; ++j) {
      v8f c = {0.f, 0.f, 0.f, 0.f, 0.f, 0.f, 0.f, 0.f};
#pragma unroll
      for (int kk = 0; kk < 16; ++kk) {
        int k = kk * 4 + 2 * half;
        v2f bf;
        bf.x = Ks[(j * 16 + m) * KS_STRIDE + k];
        bf.y = Ks[(j * 16 + m) * KS_STRIDE + k + 1];
        c = wmma4(aq[kk], bf, c);
      }
#pragma unroll
      for (int r = 0; r < 8; ++r)
        Sw[(r + half * 8) * SS_STRIDE + j * 16 + m] = c[r] * 0.125f;
    }

    // ---- nested-window softmax fused with expert gates (lanes 0..15) ----
    if (lane < 16) {
      int t = strip * 16 + lane;
      if (t < TLEN) {
        float* Srow = Sw + lane * SS_STRIDE;
        float mx = -3.4e38f;
        for (int s = 0; s < TLEN; ++s) mx = fmaxf(mx, Srow[s]);
        float s243 = 0.f;
        for (int s = 0; s < TLEN; ++s) {
          float p = __expf(Srow[s] - mx);
          Srow[s] = p;
          s243 += p;
        }
        int b9 = (t / 9) * 9, b27 = (t / 27) * 27, b81 = (t / 81) * 81;
        float s9 = 0.f, s27 = 0.f, s81 = 0.f;
        for (int s = 0; s < 9; ++s) s9 += Srow[b9 + s];
        for (int s = 0; s < 27; ++s) s27 += Srow[b27 + s];
        for (int s = 0; s < 81; ++s) s81 += Srow[b81 + s];
        const float* g = Gg + ((size_t)(b * TLEN + t) * NJ + n) * 4;
        float c9   = g[0] / fmaxf(s9, 1e-30f);
        float c27  = g[1] / fmaxf(s27, 1e-30f);
        float c81  = g[2] / fmaxf(s81, 1e-30f);
        float c243 = g[3] / fmaxf(s243, 1e-30f);
        for (int s = 0; s < TLEN; ++s) {
          float coef = c243;
          if (s >= b81 && s < b81 + 81) coef += c81;
          if (s >= b27 && s < b27 + 27) coef += c27;
          if (s >= b9 && s < b9 + 9) coef += c9;
          Srow[s] *= coef;
        }
        for (int s = TLEN; s < 256; ++s) Srow[s] = 0.f;  // zero pad cols
      }
    }
    // same-wave LDS ordering guarantees visibility; no barrier needed.

    // ---- O(16x64) = W(16x256) @ V(256x64) ----
    v8f zero = {0.f, 0.f, 0.f, 0.f, 0.f, 0.f, 0.f, 0.f};
    v8f oc[4] = {zero, zero, zero, zero};
    for (int kk = 0; kk < 64; ++kk) {
      int k = kk * 4 + 2 * half;
      v2f af;
      af.x = Sw[m * SS_STRIDE + k];
      af.y = Sw[m * SS_STRIDE + k + 1];
#pragma unroll
      for (int ct = 0; ct < 4; ++ct) {
        v2f bf;
        bf.x = Vs[(k + 0) * VS_STRIDE + ct * 16 + m];
        bf.y = Vs[(k + 1) * VS_STRIDE + ct * 16 + m];
        oc[ct] = wmma4(af, bf, oc[ct]);
      }
    }
#pragma unroll
    for (int ct = 0; ct < 4; ++ct) {
#pragma unroll
      for (int r = 0; r < 8; ++r) {
        int t = strip * 16 + r + half * 8;
        if (t < TLEN) {
          Cg[((size_t)(b * TLEN + t) * NJ + n) * DIMC + h * HEADD + ct * 16 +
             m] = oc[ct][r];
        }
      }
    }
  }
}

// ---------------------------------------------------------------------------
// Kernel 4: out = ctx @ proj_w + proj_b  (16524 x 512 x 512)
// ---------------------------------------------------------------------------
__global__ __launch_bounds__(256) void gemm_proj_kernel(
    const float* __restrict__ X, const float* __restrict__ W,
    const float* __restrict__ bias, float* __restrict__ out) {
  __shared__ float As[16][128];
  __shared__ float Bs[16][64];
  const int tid  = threadIdx.x;
  const int lane = tid & 31;
  const int wv   = tid >> 5;
  const int m    = lane & 15;
  const int half = lane >> 4;
  const int rowBase = blockIdx.y * 128;
  const int colBase = blockIdx.x * 64;

  v8f zero = {0.f, 0.f, 0.f, 0.f, 0.f, 0.f, 0.f, 0.f};
  v8f acc[4] = {zero, zero, zero, zero};

  for (int k0 = 0; k0 < DIMC; k0 += 16) {
    __syncthreads();
    {
      int r  = tid >> 1;
      int kk = (tid & 1) * 8;
      int rg = rowBase + r;
      float4 v0 = {0.f, 0.f, 0.f, 0.f}, v1 = {0.f, 0.f, 0.f, 0.f};
      if (rg < NTOK) {
        const float4* p = (const float4*)(X + (size_t)rg * DIMC + k0 + kk);
        v0 = p[0];
        v1 = p[1];
      }
      float tmp[8] = {v0.x, v0.y, v0.z, v0.w, v1.x, v1.y, v1.z, v1.w};
#pragma unroll
      for (int j = 0; j < 8; ++j) As[kk + j][r] = tmp[j];
    }
    {
      int kr = tid >> 4;
      int c  = (tid & 15) * 4;
      float4 v = *(const float4*)(W + (size_t)(k0 + kr) * DIMC + colBase + c);
      *(float4*)&Bs[kr][c] = v;
    }
    __syncthreads();
#pragma unroll
    for (int kk = 0; kk < 16; kk += 4) {
      v2f a;
      a.x = As[kk + 2 * half + 0][wv * 16 + m];
      a.y = As[kk + 2 * half + 1][wv * 16 + m];
#pragma unroll
      for (int ct = 0; ct < 4; ++ct) {
        v2f b;
        b.x = Bs[kk + 2 * half + 0][ct * 16 + m];
        b.y = Bs[kk + 2 * half + 1][ct * 16 + m];
        acc[ct] = wmma4(a, b, acc[ct]);
      }
    }
  }

#pragma unroll
  for (int ct = 0; ct < 4; ++ct) {
    int ncol = colBase + ct * 16 + m;
    float pb = bias[ncol];
#pragma unroll
    for (int r = 0; r < 8; ++r) {
      int rg = rowBase + wv * 16 + r + half * 8;
      if (rg < NTOK) out[(size_t)rg * DIMC + ncol] = acc[ct][r] + pb;
    }
  }
}

// ---------------------------------------------------------------------------
extern "C" void kernel_launch(void* const* d_in, const int* in_sizes, int n_in,
                              void* d_out, int out_size, void* d_ws,
                              size_t ws_size, hipStream_t stream) {
  (void)in_sizes; (void)n_in; (void)out_size; (void)ws_size;
  const float* x      = (const float*)d_in[0];
  const float* qkv_w  = (const float*)d_in[1];
  const float* proj_w = (const float*)d_in[2];
  const float* proj_b = (const float*)d_in[3];
  const float* te_w   = (const float*)d_in[4];
  const float* te_b   = (const float*)d_in[5];
  float* ws  = (float*)d_ws;
  float* out = (float*)d_out;

  dim3 blk(256);
  gemm_qkv_kernel<<<dim3(1536 / 64, (NTOK + 127) / 128), blk, 0, stream>>>(
      x, qkv_w, ws);
  gate_kernel<<<dim3((NTOK + 7) / 8), blk, 0, stream>>>(x, te_w, te_b,
                                                        ws + G_OFF);
  size_t lds_bytes =
      (size_t)(256 * KS_STRIDE + 256 * VS_STRIDE + 8 * 16 * SS_STRIDE) *
      sizeof(float);  // 268,288 B < 320 KB WGP LDS
  attn_kernel<<<dim3(NSEQ), blk, lds_bytes, stream>>>(ws);
  gemm_proj_kernel<<<dim3(DIMC / 64, (NTOK + 127) / 128), blk, 0, stream>>>(
      ws + CTX_OFF, proj_w, proj_b, out);
}